// AVWGCN_47064251629611
// MI455X (gfx1250) — compile-verified
//
#include <hip/hip_runtime.h>
#include <math.h>

// ---------------------------------------------------------------------------
// AVWGCN for MI455X (gfx1250, wave32).
//
// Math: out = einsum('bnki,nkio->bno', x_g, E@Wpool) + E@Bpool
//   with x_g built from supports [I, A, 2A^2 - I], A = softmax(relu(E E^T)).
// We use T2@X = 2*A*(A@X) - X to avoid the O(N^3) A@A (137 GFLOP -> ~5 GFLOP).
// A (64MB) is never materialized: generated on the fly inside the WMMA GEMM
// from E (160KB, cached in LDS) + per-row softmax stats.
// GEMM runs on V_WMMA_F32_16X16X4_F32 (exact f32, matches reference dtype).
// ---------------------------------------------------------------------------

typedef __attribute__((ext_vector_type(2))) float v2f;
typedef __attribute__((ext_vector_type(8))) float v8f;

#define N_NODES 4096
#define DIMD    10          // embedding dim
#define BC      64          // B*C = 4*16 columns of the GEMM
#define ROWS_PER_BLK 16     // one 16-row WMMA strip per block
#define WAVES   4           // 4 waves split K four ways
#define KSPLIT  (N_NODES / WAVES)   // 1024 K per wave

// ---------------------------------------------------------------------------
// Kernel 1: per-row softmax stats over relu(E E^T): rowmax[n], rinv[n]=1/sum.
// ---------------------------------------------------------------------------
__global__ void row_stats_kernel(const float* __restrict__ E,
                                 float* __restrict__ rowmax,
                                 float* __restrict__ rinv) {
  const int n   = blockIdx.x;
  const int tid = threadIdx.x;
  __shared__ float sred[256];
  __shared__ float sen[DIMD];
  if (tid < DIMD) sen[tid] = E[n * DIMD + tid];
  __syncthreads();
  float en[DIMD];
#pragma unroll
  for (int d = 0; d < DIMD; ++d) en[d] = sen[d];

  // pass 1: row max of relu(e_n . e_m)  (relu => max is >= 0)
  float mloc = 0.0f;
  for (int m = tid; m < N_NODES; m += 256) {
    float z = 0.0f;
#pragma unroll
    for (int d = 0; d < DIMD; ++d) z += en[d] * E[m * DIMD + d];
    mloc = fmaxf(mloc, fmaxf(z, 0.0f));
  }
  sred[tid] = mloc;
  __syncthreads();
  for (int s = 128; s > 0; s >>= 1) {
    if (tid < s) sred[tid] = fmaxf(sred[tid], sred[tid + s]);
    __syncthreads();
  }
  const float rmax = sred[0];
  __syncthreads();

  // pass 2: sum exp(z - rmax)
  float ssum = 0.0f;
  for (int m = tid; m < N_NODES; m += 256) {
    float z = 0.0f;
#pragma unroll
    for (int d = 0; d < DIMD; ++d) z += en[d] * E[m * DIMD + d];
    ssum += __expf(fmaxf(z, 0.0f) - rmax);
  }
  sred[tid] = ssum;
  __syncthreads();
  for (int s = 128; s > 0; s >>= 1) {
    if (tid < s) sred[tid] += sred[tid + s];
    __syncthreads();
  }
  if (tid == 0) {
    rowmax[n] = rmax;
    rinv[n]   = 1.0f / sred[0];
  }
}

// ---------------------------------------------------------------------------
// Kernel 2: x[B,N,C] -> Xt[N, B*C]  (GEMM-friendly layout, col = b*16+c)
// ---------------------------------------------------------------------------
__global__ void transpose_x_kernel(const float* __restrict__ x,
                                   float* __restrict__ Xt) {
  const int t = blockIdx.x * blockDim.x + threadIdx.x;
  if (t >= N_NODES * BC) return;
  const int n  = t / BC;
  const int bc = t % BC;
  const int b  = bc >> 4;
  const int c  = bc & 15;
  Xt[t] = x[(b * N_NODES + n) * 16 + c];
}

// ---------------------------------------------------------------------------
// Kernel 3: Y[N,64] = A @ Xin[N,64], A generated on the fly.
//   grid = 256 blocks x 128 threads (4 waves). Block owns 16 output rows,
//   each wave accumulates a K-quarter with v_wmma_f32_16x16x4_f32, then a
//   cross-wave LDS reduction. E lives entirely in LDS (160 KB of 320 KB/WGP).
//
// ISA 16x16x4 f32 fragment layout (CDNA5 ISA):
//   A (16x4): lane l = 16*half + r holds A[r][2*half + j] in component j.
//   B (4x16): lane l holds B[2*half + j][r].
//   C (16x16): component v at lane l is C[v + 8*half][r].
// ---------------------------------------------------------------------------
__global__ void agemm_kernel(const float* __restrict__ E,
                             const float* __restrict__ rowmax,
                             const float* __restrict__ rinv,
                             const float* __restrict__ Xin,
                             float* __restrict__ Yout) {
  __shared__ float sE[N_NODES * DIMD];        // 160 KB: all of E
  __shared__ float sAcc[WAVES][4][32][8];     // 16 KB: cross-wave reduction

  const int tid  = threadIdx.x;
  const int wave = tid >> 5;
  const int lane = tid & 31;
  const int half = lane >> 4;
  const int r    = lane & 15;
  const int n0   = blockIdx.x * ROWS_PER_BLK;

  // stage E into LDS as float4 (40960 floats == 10240 exactly-aligned b128)
  {
    const float4* __restrict__ E4 = (const float4*)E;
    float4* __restrict__ sE4      = (float4*)sE;
    for (int i = tid; i < (N_NODES * DIMD) / 4; i += WAVES * 32) sE4[i] = E4[i];
  }
  __syncthreads();

  // this lane's output row and its softmax stats
  const int myrow = n0 + r;
  float en[DIMD];
#pragma unroll
  for (int d = 0; d < DIMD; ++d) en[d] = sE[myrow * DIMD + d];
  const float rm = rowmax[myrow];
  const float ri = rinv[myrow];

  v8f acc[4] = {};  // 4 column fragments (cols cf*16 .. cf*16+15)

  const int kbase = wave * KSPLIT;
#pragma unroll 2
  for (int kk = 0; kk < KSPLIT; kk += 4) {
    const int k0 = kbase + kk + 2 * half;

    // branchless prefetch of Xin rows two steps ahead (global_prefetch_b8);
    // index wraps inside this wave's K range so it is always a valid address.
    __builtin_prefetch(&Xin[(kbase + ((kk + 8) & (KSPLIT - 1))) * BC + lane], 0, 1);

    // generate this lane's two A entries: A[myrow][k0], A[myrow][k0+1]
    float z0 = 0.0f, z1 = 0.0f;
#pragma unroll
    for (int d = 0; d < DIMD; ++d) {
      z0 += en[d] * sE[(k0)     * DIMD + d];
      z1 += en[d] * sE[(k0 + 1) * DIMD + d];
    }
    v2f a;
    a.x = __expf(fmaxf(z0, 0.0f) - rm) * ri;
    a.y = __expf(fmaxf(z1, 0.0f) - rm) * ri;

#pragma unroll
    for (int cf = 0; cf < 4; ++cf) {
      v2f b;
      b.x = Xin[(k0)     * BC + cf * 16 + r];
      b.y = Xin[(k0 + 1) * BC + cf * 16 + r];
      acc[cf] = __builtin_amdgcn_wmma_f32_16x16x4_f32(
          /*neg_a=*/false, a, /*neg_b=*/false, b,
          /*c_mod=*/(short)0, acc[cf], /*reuse_a=*/false, /*reuse_b=*/false);
    }
  }

  // cross-wave K reduction through LDS
#pragma unroll
  for (int cf = 0; cf < 4; ++cf)
#pragma unroll
    for (int v = 0; v < 8; ++v) sAcc[wave][cf][lane][v] = acc[cf][v];
  __syncthreads();

  {
    const int cf = wave;  // wave w finalizes column fragment w
#pragma unroll
    for (int v = 0; v < 8; ++v) {
      const float s = sAcc[0][cf][lane][v] + sAcc[1][cf][lane][v] +
                      sAcc[2][cf][lane][v] + sAcc[3][cf][lane][v];
      const int row = v + 8 * half;
      Yout[(n0 + row) * BC + cf * 16 + r] = s;
    }
  }
}

// ---------------------------------------------------------------------------
// Kernel 4: hypernetwork epilogue. One block (64 threads) per node n.
//   W_n[k,i,o] = sum_d E[n,d] * Wpool[d,k,i,o]   (48x16 in LDS)
//   out[b,n,o] = sum_{k,i} xg[b,n,k,i] * W_n[k,i,o] + E[n]·Bpool[:,o]
//   xg(k=0)=x, xg(k=1)=Y1, xg(k=2)=2*Y2 - x   (Chebyshev identity)
// ---------------------------------------------------------------------------
__global__ void output_kernel(const float* __restrict__ x,
                              const float* __restrict__ E,
                              const float* __restrict__ Wp,  // [10,3,16,16]
                              const float* __restrict__ Bp,  // [10,16]
                              const float* __restrict__ Y1,
                              const float* __restrict__ Y2,
                              float* __restrict__ out) {     // [4,4096,16]
  const int n   = blockIdx.x;
  const int tid = threadIdx.x;  // 64 threads
  __shared__ float sW[48 * 16];
  __shared__ float se[DIMD];
  if (tid < DIMD) se[tid] = E[n * DIMD + tid];
  __syncthreads();

  for (int j = tid; j < 48 * 16; j += 64) {
    float s = 0.0f;
#pragma unroll
    for (int d = 0; d < DIMD; ++d) s += se[d] * Wp[d * 768 + j];
    sW[j] = s;
  }
  __syncthreads();

  const int b = tid >> 4;
  const int o = tid & 15;
  float s = 0.0f;
#pragma unroll
  for (int d = 0; d < DIMD; ++d) s += se[d] * Bp[d * 16 + o];

#pragma unroll
  for (int i = 0; i < 16; ++i) {
    const float x0 = x[(b * N_NODES + n) * 16 + i];
    const float y1 = Y1[n * BC + b * 16 + i];
    const float y2 = 2.0f * Y2[n * BC + b * 16 + i] - x0;  // (2A^2 - I)X
    s += x0 * sW[(0 * 16 + i) * 16 + o] +
         y1 * sW[(1 * 16 + i) * 16 + o] +
         y2 * sW[(2 * 16 + i) * 16 + o];
  }
  out[(b * N_NODES + n) * 16 + o] = s;
}

// ---------------------------------------------------------------------------
extern "C" void kernel_launch(void* const* d_in, const int* in_sizes, int n_in,
                              void* d_out, int out_size, void* d_ws, size_t ws_size,
                              hipStream_t stream) {
  const float* x  = (const float*)d_in[0];  // [4,4096,16]
  const float* E  = (const float*)d_in[1];  // [4096,10]
  // d_in[2] = adjs (unused)
  const float* Wp = (const float*)d_in[3];  // [10,3,16,16]
  const float* Bp = (const float*)d_in[4];  // [10,16]
  float* out = (float*)d_out;               // [4,4096,16]

  float* ws     = (float*)d_ws;
  float* Xt     = ws;                        // [4096,64]
  float* Y1     = ws + N_NODES * BC;         // [4096,64]
  float* Y2     = ws + 2 * N_NODES * BC;     // [4096,64]
  float* rowmax = ws + 3 * N_NODES * BC;     // [4096]
  float* rinv   = rowmax + N_NODES;          // [4096]

  row_stats_kernel<<<N_NODES, 256, 0, stream>>>(E, rowmax, rinv);
  transpose_x_kernel<<<(N_NODES * BC + 255) / 256, 256, 0, stream>>>(x, Xt);
  agemm_kernel<<<N_NODES / ROWS_PER_BLK, WAVES * 32, 0, stream>>>(
      E, rowmax, rinv, Xt, Y1);   // Y1 = A @ X
  agemm_kernel<<<N_NODES / ROWS_PER_BLK, WAVES * 32, 0, stream>>>(
      E, rowmax, rinv, Y1, Y2);   // Y2 = A @ (A @ X)
  output_kernel<<<N_NODES, 64, 0, stream>>>(x, E, Wp, Bp, Y1, Y2, out);
}